// ResampleBlock_39281770889911
// MI455X (gfx1250) — compile-verified
//
#include <hip/hip_runtime.h>
#include <stdint.h>

typedef float f4v __attribute__((ext_vector_type(4)));

#define B_   16
#define N_   8192
#define C_   256
#define H_   128
#define W_   128
#define HW_  (H_*W_)
#define EPS_ 1e-6f

// 3x3 Gaussian, sigma=2 -> separable 1D weights exp(-d^2/8), normalized.
constexpr double KW_ = 0.88249690258459546;   // exp(-1/8)
constexpr double KS_ = KW_ + 1.0 + KW_;
constexpr float  GHE = (float)(KW_ / KS_);    // edge tap
constexpr float  GHC = (float)(1.0 / KS_);    // center tap

// ---------------------------------------------------------------- scatter ---
__global__ void __launch_bounds__(256) k_scatter(const float* __restrict__ x,
                                                 const float* __restrict__ loc,
                                                 float* __restrict__ feat,
                                                 float* __restrict__ cnt) {
    int t    = blockIdx.x * 8 + (threadIdx.x >> 5);   // one wave32 per token
    int lane = threadIdx.x & 31;
    float lx = loc[2 * (size_t)t];
    float ly = loc[2 * (size_t)t + 1];
    lx = fminf(fmaxf(lx, -1.f), 1.f);
    ly = fminf(fmaxf(ly, -1.f), 1.f);
    int px = (int)rintf(0.5f * (lx + 1.f) * (float)W_ - 0.5f);  // round half-even
    int py = (int)rintf(0.5f * (ly + 1.f) * (float)H_ - 0.5f);
    px = min(max(px, 0), W_ - 1);
    py = min(max(py, 0), H_ - 1);
    int b = t >> 13;                                   // t / N_
    size_t pix   = (size_t)b * HW_ + py * W_ + px;
    size_t pbase = pix * C_;
    size_t xbase = (size_t)t * C_;
#pragma unroll
    for (int j = 0; j < 8; ++j) {
        int c = lane + 32 * j;
        atomicAdd(&feat[pbase + c], x[xbase + c]);     // no-return f32 atomic
    }
    if (lane == 0) atomicAdd(&cnt[pix], 1.0f);
}

// -------------------------------------------------------------- normalize ---
__global__ void __launch_bounds__(256) k_normalize(float* __restrict__ feat,
                                                   const float* __restrict__ cnt,
                                                   float* __restrict__ mask) {
    size_t i   = (size_t)blockIdx.x * 256 + threadIdx.x;  // float4 units
    size_t pix = i >> 6;
    int    c4  = (int)(i & 63);
    float  c   = cnt[pix];
    float  m   = (c > 0.f) ? 1.f : 0.f;
    float  r   = m / (c + EPS_);
    f4v v = ((f4v*)feat)[i];
    v *= r;
    ((f4v*)feat)[i] = v;
    if (c4 == 0) mask[pix] = m;
}

// ----------------------------------------------------- conv of mask plane ---
__global__ void __launch_bounds__(256) k_conv_mask(const float* __restrict__ mask,
                                                   float* __restrict__ convM) {
    int i = blockIdx.x * 256 + threadIdx.x;            // B*HW pixels
    int b = i >> 14;
    int p = i & (HW_ - 1);
    int y = p >> 7, x = p & 127;
    const float* mb = mask + (size_t)b * HW_;
    const float w[3] = {GHE, GHC, GHE};
    float s = 0.f;
#pragma unroll
    for (int j = 0; j < 3; ++j) {
        int yy = y + j - 1;
        if (yy < 0 || yy >= H_) continue;
#pragma unroll
        for (int k = 0; k < 3; ++k) {
            int xx = x + k - 1;
            if (xx < 0 || xx >= W_) continue;
            s += w[j] * w[k] * mb[yy * W_ + xx];
        }
    }
    convM[i] = s;
}

// ------------------------------- conv of feat + masked fill (LDS staging) ---
// Block: 16x16 pixel tile x 32 channels. Halo tile 18x18x32 staged into LDS
// with gfx1250 async global->LDS (b128), synced with s_wait_asynccnt.
__global__ void __launch_bounds__(256) k_conv_feat(const float* __restrict__ feat,
                                                   const float* __restrict__ mask,
                                                   const float* __restrict__ convM,
                                                   float* __restrict__ fmap) {
    __shared__ f4v lds4[18 * 18 * 8];                  // 41472 B
    float* lds = (float*)lds4;

    int id = blockIdx.x;                               // B * 8 * 8 * 8 blocks
    int cc = id & 7;
    int tx = (id >> 3) & 7;
    int ty = (id >> 6) & 7;
    int b  = id >> 9;
    int c0 = cc * 32, x0 = tx * 16, y0 = ty * 16;
    int tid = threadIdx.x;

    uint32_t lbase = (uint32_t)(uintptr_t)lds;         // LDS byte offset (addr[31:0])
    const float* fb = feat + (size_t)b * HW_ * C_;

    for (int u = tid; u < 18 * 18 * 8; u += 256) {
        int p  = u >> 3;                               // tile pixel
        int f4 = u & 7;                                // float4 within 32-ch chunk
        int py = p / 18;
        int pxl = p - py * 18;
        int gy = y0 - 1 + py;
        int gx = x0 - 1 + pxl;
        uint32_t loff = lbase + (uint32_t)(u * 16);
        if (gy >= 0 && gy < H_ && gx >= 0 && gx < W_) {
            const float* gp = fb + ((size_t)(gy * W_ + gx)) * C_ + c0 + f4 * 4;
            asm volatile("global_load_async_to_lds_b128 %0, %1, off"
                         :: "v"(loff), "v"((unsigned long long)(uintptr_t)gp)
                         : "memory");
        } else {
            f4v z = {0.f, 0.f, 0.f, 0.f};              // zero-pad SAME halo
            lds4[u] = z;
        }
    }
    asm volatile("s_wait_asynccnt 0x0" ::: "memory");  // async LDS writes landed
    __syncthreads();

    int l = tid & 31;                                  // channel within chunk
    int r = tid >> 5;                                  // row group 0..7
    const float* __restrict__ L = lds;
    auto colv = [&](int yy, int xx) -> float {         // vertical 3-tap on column
        int bi = (yy * 18 + xx) * 32 + l;
        return GHE * L[bi] + GHC * L[bi + 18 * 32] + GHE * L[bi + 2 * 18 * 32];
    };
#pragma unroll
    for (int yy = r; yy < 16; yy += 8) {
        float cA = colv(yy, 0);
        float cB = colv(yy, 1);
        for (int xx = 0; xx < 16; ++xx) {              // slide horizontal 3-tap
            float cC   = colv(yy, xx + 2);
            float conv = GHE * cA + GHC * cB + GHE * cC;
            float ctr  = L[((yy + 1) * 18 + (xx + 1)) * 32 + l];
            size_t pidx = (size_t)b * HW_ + (y0 + yy) * W_ + (x0 + xx);
            float m  = mask[pidx];
            float cM = convM[pidx];
            float fi = conv / (cM + EPS_);
            fi = (cM > 0.f) ? fi : 0.f;
            fmap[pidx * C_ + c0 + l] = ctr + (1.f - m) * fi;
            cA = cB; cB = cC;
        }
    }
}

// ------------------------------------------------------- bilinear gather ----
__global__ void __launch_bounds__(256) k_gather(const float* __restrict__ fmap,
                                                const float* __restrict__ loc_out,
                                                float* __restrict__ out) {
    size_t unit = (size_t)blockIdx.x * 256 + threadIdx.x;  // B*N*64 float4 units
    int t  = (int)(unit >> 6);
    int c4 = (int)(unit & 63);
    int b  = t >> 13;
    float lx = loc_out[2 * (size_t)t];
    float ly = loc_out[2 * (size_t)t + 1];
    float ix = ((lx + 1.f) * (float)W_ - 1.f) * 0.5f;
    float iy = ((ly + 1.f) * (float)H_ - 1.f) * 0.5f;
    float xf = floorf(ix), yf = floorf(iy);
    float wx = ix - xf, wy = iy - yf;
    int x0i = (int)xf, y0i = (int)yf;

    const f4v* f4 = (const f4v*)(fmap + (size_t)b * HW_ * C_);
    f4v acc = {0.f, 0.f, 0.f, 0.f};
    auto corner = [&](int xi, int yi, float w) {
        float valid = (xi >= 0 && xi < W_ && yi >= 0 && yi < H_) ? 1.f : 0.f;
        int xc = min(max(xi, 0), W_ - 1);
        int yc = min(max(yi, 0), H_ - 1);
        f4v v = f4[(size_t)(yc * W_ + xc) * 64 + c4];
        acc += (w * valid) * v;
    };
    corner(x0i,     y0i,     (1.f - wx) * (1.f - wy));
    corner(x0i + 1, y0i,     wx * (1.f - wy));
    corner(x0i,     y0i + 1, (1.f - wx) * wy);
    corner(x0i + 1, y0i + 1, wx * wy);

    __builtin_nontemporal_store(acc, &((f4v*)out)[unit]);  // stream 134 MB out
}

// ------------------------------------------------------------------ launch --
extern "C" void kernel_launch(void* const* d_in, const int* in_sizes, int n_in,
                              void* d_out, int out_size, void* d_ws, size_t ws_size,
                              hipStream_t stream) {
    (void)in_sizes; (void)n_in; (void)out_size; (void)ws_size;
    const float* x       = (const float*)d_in[0];
    const float* loc     = (const float*)d_in[1];
    const float* loc_out = (const float*)d_in[2];

    const size_t FEAT_BYTES = (size_t)B_ * HW_ * C_ * 4;  // 256 MiB
    const size_t CNT_BYTES  = (size_t)B_ * HW_ * 4;       // 1 MiB
    char* ws = (char*)d_ws;
    float* feat  = (float*)(ws);
    float* fmap  = (float*)(ws + FEAT_BYTES);
    float* cnt   = (float*)(ws + 2 * FEAT_BYTES);
    float* mask  = (float*)(ws + 2 * FEAT_BYTES + CNT_BYTES);
    float* convM = (float*)(ws + 2 * FEAT_BYTES + 2 * CNT_BYTES);

    hipMemsetAsync(feat, 0, FEAT_BYTES, stream);
    hipMemsetAsync(cnt, 0, CNT_BYTES, stream);

    k_scatter  <<<B_ * N_ / 8,             256, 0, stream>>>(x, loc, feat, cnt);
    k_normalize<<<(B_ * HW_ * (C_/4))/256, 256, 0, stream>>>(feat, cnt, mask);
    k_conv_mask<<<B_ * HW_ / 256,          256, 0, stream>>>(mask, convM);
    k_conv_feat<<<B_ * 8 * 8 * 8,          256, 0, stream>>>(feat, mask, convM, fmap);
    k_gather   <<<B_ * N_ * 64 / 256,      256, 0, stream>>>(fmap, loc_out, (float*)d_out);
}